// TenGCN_78623671321103
// MI455X (gfx1250) — compile-verified
//
#include <hip/hip_runtime.h>

#define DEVINL __device__ __forceinline__

namespace {

constexpr int B  = 64;
constexpr int N  = 1024;
constexpr int E  = 32768;
constexpr int F  = 128;
constexpr int H  = 32;
constexpr int K  = 512;
constexpr int NB = 5;
constexpr int PH = 25;     // P/2
constexpr int OUT = 2;
constexpr int HS = 34;     // padded LDS row stride (bank-conflict-free, 8B aligned)

// const-block layout inside workspace (float offsets)
constexpr int WFUSE = 0;      // 32x32
constexpr int BFUSE = 1024;   // 32
constexpr int F2V   = 1056;   // 32
constexpr int G2V   = 1088;   // 25 (pad 32)
constexpr int ALPHA = 1120;   // 2  (pad 8)
constexpr int BETA  = 1128;   // 32
constexpr int CSTN  = 2048;   // floats reserved

typedef float v2f __attribute__((ext_vector_type(2)));
typedef float v8f __attribute__((ext_vector_type(8)));

DEVINL v8f wmma4(v2f a, v2f b, v8f c) {
  // V_WMMA_F32_16X16X4_F32 : D = A(16x4,f32) * B(4x16,f32) + C(16x16,f32)
  return __builtin_amdgcn_wmma_f32_16x16x4_f32(false, a, false, b, (short)0, c,
                                               false, false);
}

// ---------------------------------------------------------------------------
// Phase 0: fold all tail linear algebra into tiny constants + Wfuse.
// ---------------------------------------------------------------------------
__global__ __launch_bounds__(256) void precompute_kernel(
    const float* __restrict__ Wm0b, const float* __restrict__ Wg1,
    const float* __restrict__ bm0b, const float* __restrict__ H0,
    const float* __restrict__ H2,   const float* __restrict__ F2,
    const float* __restrict__ G2,   const float* __restrict__ F0,
    const float* __restrict__ G0,   float* __restrict__ cst) {
  __shared__ float hcolL[64];
  __shared__ float h2sL[32];
  const int t = threadIdx.x;

  if (t < 64) {                 // hcol[i] = sum_p H0[p,i]
    float s = 0.f;
    for (int p = 0; p < 64; ++p) s += H0[p * 64 + t];
    hcolL[t] = s;
  } else if (t < 96) {          // h2s[k] = sum_r H2[r,k]
    const int k = t - 64;
    float s = 0.f;
    for (int r = 0; r < 32; ++r) s += H2[r * 32 + k];
    h2sL[k] = s;
  }
  __syncthreads();

  if (t < 32) {                 // f2v = F2^T h2s
    float s = 0.f;
    for (int k = 0; k < 32; ++k) s += F2[k * 32 + t] * h2sL[k];
    cst[F2V + t] = s;
  } else if (t < 64) {          // g2v = G2^T h2s
    const int k = t - 32;
    if (k < PH) {
      float s = 0.f;
      for (int r = 0; r < 32; ++r) s += G2[r * PH + k] * h2sL[r];
      cst[G2V + k] = s;
    }
  } else if (t < 66) {          // alpha_l = sum_{i<32} hcol[i] F0[i,l]
    const int l = t - 64;
    float s = 0.f;
    for (int i = 0; i < 32; ++i) s += hcolL[i] * F0[i * 2 + l];
    cst[ALPHA + l] = s;
  } else if (t < 98) {          // beta_o = sum_m hcol[32+m] G0[m,o]
    const int o = t - 66;
    float s = 0.f;
    for (int m = 0; m < 32; ++m) s += hcolL[32 + m] * G0[m * 32 + o];
    cst[BETA + o] = s;
  }

  // Wfuse = Wm0b @ W_gcn1 (32x1024 @ 1024x32), bfuse = bm0b @ W_gcn1
  for (int o = t; o < 1024; o += 256) {
    const int j = o >> 5, n = o & 31;
    float s = 0.f;
    for (int u = 0; u < 1024; ++u) s += Wm0b[j * 1024 + u] * Wg1[u * 32 + n];
    cst[WFUSE + o] = s;
  }
  if (t < 32) {
    float s = 0.f;
    for (int u = 0; u < 1024; ++u) s += bm0b[u] * Wg1[u * 32 + t];
    cst[BFUSE + t] = s;
  }
}

// ---------------------------------------------------------------------------
// Score: emb = gcn_conv(x, W_score); score = sum_nb |emb pair diff|.
// Only the 5 paired differences matter, so project through
// dW[:,t] = W_score[:,2t+1]-W_score[:,2t] (5 dims instead of 10).
// One workgroup per graph; degree + aggregation via LDS float atomics.
// ---------------------------------------------------------------------------
__global__ __launch_bounds__(256) void score_kernel(
    const float* __restrict__ x, const int* __restrict__ ei,
    const float* __restrict__ Ws, const float* __restrict__ bs,
    float* __restrict__ score) {
  __shared__ float dinvL[N];        // degree, then rsqrt
  __shared__ float hd[N * NB];
  __shared__ float agd[N * NB];
  const int b = blockIdx.x, tid = threadIdx.x;
  const int* srcI = ei + (size_t)b * 2 * E;
  const int* dstI = srcI + E;
  const float* xb = x + (size_t)b * N * F;

  for (int i = tid; i < N; i += 256) dinvL[i] = 0.f;
  for (int i = tid; i < N * NB; i += 256) agd[i] = 0.f;
  __syncthreads();

  for (int e = tid; e < E; e += 256) atomicAdd(&dinvL[dstI[e]], 1.0f);

  for (int n = tid; n < N; n += 256) {
    float acc[NB] = {0.f, 0.f, 0.f, 0.f, 0.f};
    const float* xr = xb + n * F;
    for (int f = 0; f < F; ++f) {
      const float xv = xr[f];
#pragma unroll
      for (int t = 0; t < NB; ++t)
        acc[t] += xv * (Ws[f * (2 * NB) + 2 * t + 1] - Ws[f * (2 * NB) + 2 * t]);
    }
#pragma unroll
    for (int t = 0; t < NB; ++t) hd[n * NB + t] = acc[t];
  }
  __syncthreads();

  for (int i = tid; i < N; i += 256) dinvL[i] = rsqrtf(dinvL[i] + 1.0f);
  __syncthreads();

  for (int e = tid; e < E; e += 256) {
    const int s = srcI[e], d = dstI[e];
    const float coef = dinvL[s] * dinvL[d];
#pragma unroll
    for (int t = 0; t < NB; ++t) atomicAdd(&agd[d * NB + t], hd[s * NB + t] * coef);
  }
  __syncthreads();

  for (int n = tid; n < N; n += 256) {
    const float di2 = dinvL[n] * dinvL[n];
    float sc = 0.f;
#pragma unroll
    for (int t = 0; t < NB; ++t)
      sc += fabsf(agd[n * NB + t] + hd[n * NB + t] * di2 +
                  (bs[2 * t + 1] - bs[2 * t]));
    score[(size_t)b * N + n] = sc;
  }
}

// ---------------------------------------------------------------------------
// Top-K (K=512 of 1024): full bitonic sort in LDS; ties -> lower index first
// (matches jax.lax.top_k). Emits perm and newidx.
// ---------------------------------------------------------------------------
__global__ __launch_bounds__(512) void topk_kernel(
    const float* __restrict__ score, int* __restrict__ perm,
    int* __restrict__ newidx) {
  __shared__ float keyL[N];
  __shared__ int idxL[N];
  const int b = blockIdx.x, tid = threadIdx.x;
  for (int i = tid; i < N; i += 512) {
    keyL[i] = -score[(size_t)b * N + i];  // ascending(-s) == descending(s)
    idxL[i] = i;
  }
  for (int k = 2; k <= N; k <<= 1) {
    for (int j = k >> 1; j > 0; j >>= 1) {
      __syncthreads();
      for (int i = tid; i < N; i += 512) {
        const int ixj = i ^ j;
        if (ixj > i) {
          const float ka = keyL[i], kb = keyL[ixj];
          const int ia = idxL[i], ib = idxL[ixj];
          const bool gt = (ka > kb) || (ka == kb && ia > ib);
          const bool up = ((i & k) == 0);
          if (gt == up) {
            keyL[i] = kb; keyL[ixj] = ka;
            idxL[i] = ib; idxL[ixj] = ia;
          }
        }
      }
    }
  }
  __syncthreads();
  for (int r = tid; r < N; r += 512) {
    const int node = idxL[r];
    newidx[(size_t)b * N + node] = (r < K) ? r : -1;
    if (r < K) perm[(size_t)b * K + r] = node;
  }
}

// ---------------------------------------------------------------------------
// Gather pooled node features directly into WMMA A-fragment order:
//   xpT[b][mt][kk][lane][h]  (mt<32 row-tiles, kk<32 k-steps, lane<32, h<2)
//   row = mt*16 + (lane&15), col = kk*4 + 2*(lane>>4) + h
// so mega's A-frag load is one contiguous 256B transaction per wave.
// ---------------------------------------------------------------------------
__global__ __launch_bounds__(256) void gather_kernel(
    const float* __restrict__ x, const int* __restrict__ perm,
    float* __restrict__ xpT) {
  const size_t i = (size_t)blockIdx.x * 256 + threadIdx.x;  // < B*K*F
  const int h    = (int)(i & 1);
  const int lane = (int)((i >> 1) & 31);
  const int kk   = (int)((i >> 6) & 31);   // F/4 = 32
  const int mt   = (int)((i >> 11) & 31);  // K/16 = 32
  const int b    = (int)(i >> 16);
  const int row  = mt * 16 + (lane & 15);
  const int col  = kk * 4 + 2 * (lane >> 4) + h;
  const int node = perm[(size_t)b * K + row];
  xpT[i] = x[((size_t)b * N + node) * F + col];
}

__global__ __launch_bounds__(256) void remap_kernel(
    const int* __restrict__ ei, const int* __restrict__ newidx,
    int* __restrict__ ps, int* __restrict__ pd, float* __restrict__ em) {
  const size_t i = (size_t)blockIdx.x * 256 + threadIdx.x;  // < B*E
  const int e = (int)(i & (E - 1));
  const int b = (int)(i >> 15);
  const int* base = ei + (size_t)b * 2 * E;
  const int r = newidx[(size_t)b * N + base[e]];
  const int c = newidx[(size_t)b * N + base[E + e]];
  const bool valid = (r >= 0) && (c >= 0);
  ps[i] = valid ? r : 0;
  pd[i] = valid ? c : 0;
  em[i] = valid ? 1.f : 0.f;
}

// ---------------------------------------------------------------------------
// WMMA GEMM over LDS: D(512x32) = act(A(512x32) @ W(32x32) + bias)
// 8 waves * 8 tiles, K-loop of 8 V_WMMA_F32_16X16X4_F32 each.
// LDS rows padded to HS=34 floats -> conflict-free ds_load_b64 frags.
// ---------------------------------------------------------------------------
DEVINL void gemm_tile_lds(const float* __restrict__ aL,
                          const float* __restrict__ Wg,
                          const float* __restrict__ bias, bool doRelu,
                          float* __restrict__ dL, int lane, int wv) {
  const int lm = lane & 15;
  const int kg = (lane >> 4) * 2;
  for (int t = wv; t < 64; t += 8) {
    const int mt = t >> 1, nt = t & 1;
    const int row = mt * 16 + lm;
    const int col = nt * 16 + lm;
    v8f acc = {};
#pragma unroll
    for (int k0 = 0; k0 < H; k0 += 4) {
      const v2f a = *(const v2f*)(aL + row * HS + k0 + kg);
      v2f w;
      w.x = Wg[(k0 + kg) * H + col];
      w.y = Wg[(k0 + kg + 1) * H + col];
      acc = wmma4(a, w, acc);
    }
    const float bv = bias[col];
    const int rbase = mt * 16 + (lane >> 4) * 8;
#pragma unroll
    for (int v = 0; v < 8; ++v) {
      float r = acc[v] + bv;
      if (doRelu) r = fmaxf(r, 0.f);
      dL[(rbase + v) * HS + col] = r;
    }
  }
}

DEVINL void scatter_edges(const int* __restrict__ ps, const int* __restrict__ pd,
                          const float* __restrict__ emv,
                          const float* __restrict__ hsrc, float* __restrict__ agg,
                          const float* __restrict__ dinvL, int tid) {
  for (int e = tid; e < E; e += 256) {
    __builtin_prefetch(emv + e + 2048, 0, 1);
    __builtin_prefetch(ps + e + 2048, 0, 1);
    const float m = emv[e];
    if (m > 0.f) {
      const int s = ps[e], d = pd[e];
      const float coef = dinvL[s] * dinvL[d];
      const float* hs = hsrc + s * HS;
      float* ad = agg + d * HS;
#pragma unroll
      for (int j = 0; j < H; ++j) atomicAdd(ad + j, hs[j] * coef);
    }
  }
}

// ---------------------------------------------------------------------------
// Per-graph megakernel: conv0 -> t0 -> (fused conv1) -> t1, all in LDS.
// Emits r0/r1 = column sums of relu'd hidden states (node-mean * K).
// ---------------------------------------------------------------------------
__global__ __launch_bounds__(256) void mega_kernel(
    const float* __restrict__ xpT, const int* __restrict__ psrc,
    const int* __restrict__ pdst, const float* __restrict__ em,
    const float* __restrict__ Wg0, const float* __restrict__ bg0,
    const float* __restrict__ Wm0a, const float* __restrict__ bm0a,
    const float* __restrict__ cst, const float* __restrict__ bg1,
    const float* __restrict__ Wm1a, const float* __restrict__ bm1a,
    float* __restrict__ r0, float* __restrict__ r1) {
  __shared__ float hbuf[K * HS];    // ~68 KB
  __shared__ float aggbuf[K * HS];  // ~68 KB
  __shared__ float dinvL[K];
  const int b = blockIdx.x, tid = threadIdx.x;
  const int lane = tid & 31, wv = tid >> 5;
  const float* xpb = xpT + (size_t)b * K * F;
  const int* ps = psrc + (size_t)b * E;
  const int* pd = pdst + (size_t)b * E;
  const float* emv = em + (size_t)b * E;
  const float* wfuse = cst + WFUSE;
  const float* bfuse = cst + BFUSE;

  for (int i = tid; i < K; i += 256) dinvL[i] = 0.f;
  for (int i = tid; i < K * HS; i += 256) aggbuf[i] = 0.f;
  __syncthreads();

  // degree (masked) via LDS atomics
  for (int e = tid; e < E; e += 256) atomicAdd(&dinvL[pd[e]], emv[e]);

  // h0 = xp @ W_gcn0  (512x128 @ 128x32) via V_WMMA_F32_16X16X4_F32.
  // A-frags are pre-swizzled in xpT: one coalesced 256B wave load per step.
  {
    const int lm = lane & 15;
    const int kg = (lane >> 4) * 2;
    for (int t = wv; t < 64; t += 8) {
      const int mt = t >> 1, nt = t & 1;
      const int col = nt * 16 + lm;
      const float* afrag = xpb + (size_t)mt * 32 * 64 + lane * 2;
      v8f acc = {};
      for (int kk = 0; kk < 32; ++kk) {
        const v2f a = *(const v2f*)(afrag + kk * 64);
        const int k0 = kk * 4;
        v2f w;
        w.x = Wg0[(k0 + kg) * H + col];
        w.y = Wg0[(k0 + kg + 1) * H + col];
        acc = wmma4(a, w, acc);
      }
      const int rbase = mt * 16 + (lane >> 4) * 8;
#pragma unroll
      for (int v = 0; v < 8; ++v) hbuf[(rbase + v) * HS + col] = acc[v];
    }
  }
  __syncthreads();

  for (int i = tid; i < K; i += 256) dinvL[i] = rsqrtf(dinvL[i] + 1.f);
  __syncthreads();

  scatter_edges(ps, pd, emv, hbuf, aggbuf, dinvL, tid);   // agg of h0
  __syncthreads();

  for (int i = tid; i < K * H; i += 256) {                // c0
    const int n = i >> 5, j = i & 31, o = n * HS + j;
    hbuf[o] = aggbuf[o] + hbuf[o] * dinvL[n] * dinvL[n] + bg0[j];
  }
  __syncthreads();

  gemm_tile_lds(hbuf, Wm0a, bm0a, true, aggbuf, lane, wv);  // t0 (relu'd)
  __syncthreads();

  if (tid < H) {                                            // r0 = colsum(t0)
    float s = 0.f;
    for (int n = 0; n < K; ++n) s += aggbuf[n * HS + tid];
    r0[(size_t)b * H + tid] = s;
  }
  gemm_tile_lds(aggbuf, wfuse, bfuse, false, hbuf, lane, wv);  // hf = t0@Wfuse
  __syncthreads();

  for (int i = tid; i < K * HS; i += 256) aggbuf[i] = 0.f;
  __syncthreads();
  scatter_edges(ps, pd, emv, hbuf, aggbuf, dinvL, tid);   // agg of hf
  __syncthreads();

  for (int i = tid; i < K * H; i += 256) {                // c1
    const int n = i >> 5, j = i & 31, o = n * HS + j;
    hbuf[o] = aggbuf[o] + hbuf[o] * dinvL[n] * dinvL[n] + bg1[j];
  }
  __syncthreads();

  gemm_tile_lds(hbuf, Wm1a, bm1a, true, aggbuf, lane, wv);  // t1 (relu'd)
  __syncthreads();

  if (tid < H) {                                            // r1 = colsum(t1)
    float s = 0.f;
    for (int n = 0; n < K; ++n) s += aggbuf[n * HS + tid];
    r1[(size_t)b * H + tid] = s;
  }
}

// ---------------------------------------------------------------------------
// pe conv (stride-2 2x2 VALID) + relu, folded with beta into pebeta[b,25,25].
// ---------------------------------------------------------------------------
__global__ __launch_bounds__(256) void pe_kernel(
    const float* __restrict__ PI, const float* __restrict__ Kc,
    const float* __restrict__ bconv, const float* __restrict__ cst,
    float* __restrict__ pebeta) {
  const int b = blockIdx.x, tid = threadIdx.x;
  const float* PIb = PI + (size_t)b * 5 * 50 * 50;
  const float* beta = cst + BETA;
  for (int px = tid; px < PH * PH; px += 256) {
    const int y = px / PH, x = px % PH;
    float acc = 0.f;
    for (int o = 0; o < H; ++o) {
      float v = bconv[o];
#pragma unroll
      for (int c = 0; c < 5; ++c)
#pragma unroll
        for (int ky = 0; ky < 2; ++ky)
#pragma unroll
          for (int kx = 0; kx < 2; ++kx)
            v += PIb[(c * 50 + 2 * y + ky) * 50 + 2 * x + kx] *
                 Kc[((o * 5 + c) * 2 + ky) * 2 + kx];
      acc += beta[o] * fmaxf(v, 0.f);
    }
    pebeta[(size_t)b * PH * PH + px] = acc;
  }
}

// ---------------------------------------------------------------------------
// Tail: collapsed tcl/fuse/mean pipeline + output MLP. One wave per batch.
// ---------------------------------------------------------------------------
__global__ __launch_bounds__(32) void tail_kernel(
    const float* __restrict__ r0, const float* __restrict__ r1,
    const float* __restrict__ Wm0b, const float* __restrict__ bm0b,
    const float* __restrict__ Wm1b, const float* __restrict__ bm1b,
    const float* __restrict__ cst, const float* __restrict__ F1,
    const float* __restrict__ G1, const float* __restrict__ H1,
    const float* __restrict__ Wo1, const float* __restrict__ bo1,
    const float* __restrict__ Wo2, const float* __restrict__ bo2,
    const float* __restrict__ pebeta, float* __restrict__ outp) {
  __shared__ float gv[32], pv[32], sv[32], pooledL[32], av[32];
  const int b = blockIdx.x, t = threadIdx.x;
  const float* f2v = cst + F2V;
  const float* g2v = cst + G2V;
  const float a0c = cst[ALPHA + 0], a1c = cst[ALPHA + 1];
  const float inv512 = 1.f / (float)K;

  // gv[j] = sum_k G[b][j,k] * f2v[k],  G = a0*g0 + a1*g1 (32x32 reshapes)
  {
    float g = 0.f;
    for (int k2 = 0; k2 < 32; ++k2) {
      const int col = t * 32 + k2;
      float m0v = bm0b[col], m1v = bm1b[col];
      for (int u = 0; u < 32; ++u) {
        m0v += (r0[(size_t)b * H + u] * inv512) * Wm0b[u * 1024 + col];
        m1v += (r1[(size_t)b * H + u] * inv512) * Wm1b[u * 1024 + col];
      }
      g += (a0c * m0v + a1c * m1v) * f2v[k2];
    }
    gv[t] = g;
  }
  if (t < PH) {  // pv = pebeta[b] @ g2v
    float s = 0.f;
    for (int x = 0; x < PH; ++x) s += pebeta[(size_t)b * PH * PH + t * PH + x] * g2v[x];
    pv[t] = s;
  }
  __syncthreads();

  float sA = 0.f;
  for (int j = 0; j < 32; ++j) sA += F1[t * 32 + j] * gv[j];
  float sB = 0.f;
  for (int j = 0; j < PH; ++j) sB += G1[t * PH + j] * pv[j];
  sv[t] = sA + sB;
  __syncthreads();

  float p = 0.f;
  for (int j = 0; j < 32; ++j) p += H1[t * 32 + j] * sv[j];
  pooledL[t] = p * (1.f / 2048.f);   // mean over p(64) and r(32)
  __syncthreads();

  float aa = bo1[t];
  for (int q = 0; q < 32; ++q) aa += pooledL[q] * Wo1[q * 32 + t];
  av[t] = fmaxf(aa, 0.f);
  __syncthreads();

  if (t < OUT) {
    float o = bo2[t];
    for (int j = 0; j < 32; ++j) o += av[j] * Wo2[j * OUT + t];
    outp[(size_t)b * OUT + t] = o;
  }
}

}  // namespace

extern "C" void kernel_launch(void* const* d_in, const int* in_sizes, int n_in,
                              void* d_out, int out_size, void* d_ws,
                              size_t ws_size, hipStream_t stream) {
  (void)in_sizes; (void)n_in; (void)out_size; (void)ws_size;
  const float* x1   = (const float*)d_in[0];
  const int*   ei1  = (const int*)d_in[1];
  const float* PI1  = (const float*)d_in[2];
  const float* x2   = (const float*)d_in[3];
  const int*   ei2  = (const int*)d_in[4];
  const float* PI2  = (const float*)d_in[5];
  const float* Wsc  = (const float*)d_in[6];
  const float* bsc  = (const float*)d_in[7];
  const float* Wg0  = (const float*)d_in[8];
  const float* bg0  = (const float*)d_in[9];
  const float* Wm0a = (const float*)d_in[10];
  const float* bm0a = (const float*)d_in[11];
  const float* Wm0b = (const float*)d_in[12];
  const float* bm0b = (const float*)d_in[13];
  const float* Wg1  = (const float*)d_in[14];
  const float* bg1  = (const float*)d_in[15];
  const float* Wm1a = (const float*)d_in[16];
  const float* bm1a = (const float*)d_in[17];
  const float* Wm1b = (const float*)d_in[18];
  const float* bm1b = (const float*)d_in[19];
  const float* F0   = (const float*)d_in[20];
  const float* F1   = (const float*)d_in[21];
  const float* F2   = (const float*)d_in[22];
  const float* Kc   = (const float*)d_in[23];
  const float* bcv  = (const float*)d_in[24];
  const float* G0   = (const float*)d_in[25];
  const float* G1   = (const float*)d_in[26];
  const float* G2   = (const float*)d_in[27];
  const float* H0   = (const float*)d_in[28];
  const float* H1m  = (const float*)d_in[29];
  const float* H2m  = (const float*)d_in[30];
  const float* Wo1  = (const float*)d_in[31];
  const float* bo1  = (const float*)d_in[32];
  const float* Wo2  = (const float*)d_in[33];
  const float* bo2  = (const float*)d_in[34];

  char* ws = (char*)d_ws;
  size_t off = 0;
  auto take = [&](size_t bytes) {
    size_t r = off;
    off = (off + bytes + 255) & ~(size_t)255;
    return r;
  };
  float* cst   = (float*)(ws + take((size_t)CSTN * 4));
  float* score = (float*)(ws + take((size_t)B * N * 4));
  int*   perm  = (int*)  (ws + take((size_t)B * K * 4));
  int*   nidx  = (int*)  (ws + take((size_t)B * N * 4));
  float* xpT   = (float*)(ws + take((size_t)B * K * F * 4));
  int*   ps    = (int*)  (ws + take((size_t)B * E * 4));
  int*   pd    = (int*)  (ws + take((size_t)B * E * 4));
  float* em    = (float*)(ws + take((size_t)B * E * 4));
  float* r0    = (float*)(ws + take((size_t)B * H * 4));
  float* r1    = (float*)(ws + take((size_t)B * H * 4));
  float* peb   = (float*)(ws + take((size_t)B * PH * PH * 4));

  precompute_kernel<<<1, 256, 0, stream>>>(Wm0b, Wg1, bm0b, H0, H2m, F2, G2,
                                           F0, G0, cst);

  for (int br = 0; br < 2; ++br) {
    const float* x   = br ? x2 : x1;
    const int*   ei  = br ? ei2 : ei1;
    const float* PIv = br ? PI2 : PI1;

    score_kernel<<<B, 256, 0, stream>>>(x, ei, Wsc, bsc, score);
    topk_kernel<<<B, 512, 0, stream>>>(score, perm, nidx);
    gather_kernel<<<(B * K * F) / 256, 256, 0, stream>>>(x, perm, xpT);
    remap_kernel<<<(B * E) / 256, 256, 0, stream>>>(ei, nidx, ps, pd, em);
    mega_kernel<<<B, 256, 0, stream>>>(xpT, ps, pd, em, Wg0, bg0, Wm0a, bm0a,
                                       cst, bg1, Wm1a, bm1a, r0, r1);
    pe_kernel<<<B, 256, 0, stream>>>(PIv, Kc, bcv, cst, peb);
    tail_kernel<<<B, 32, 0, stream>>>(r0, r1, Wm0b, bm0b, Wm1b, bm1b, cst, F1,
                                      G1, H1m, Wo1, bo1, Wo2, bo2, peb,
                                      (float*)d_out + (size_t)br * B * OUT);
  }
}